// GCN_Model_60026462929383
// MI455X (gfx1250) — compile-verified
//
#include <hip/hip_runtime.h>
#include <hip/hip_bf16.h>

typedef __attribute__((ext_vector_type(2))) float v2f;
typedef __attribute__((ext_vector_type(8))) float v8f;

#define NUM_GRAPHS 512
#define NPG 200
#define NN (NUM_GRAPHS * NPG)   /* 102400 nodes */
#define EE (NN * 16)            /* 1638400 edges */
#define EDIM 128

// ---------------------------------------------------------------------------
// utility kernels
// ---------------------------------------------------------------------------
__global__ void zero_f32(float* __restrict__ p, long n) {
    long t = (long)blockIdx.x * blockDim.x + threadIdx.x;
    if (t < n) p[t] = 0.0f;
}

__global__ void init_deg(float* __restrict__ deg) {
    int t = blockIdx.x * blockDim.x + threadIdx.x;
    if (t < NN) deg[t] = 1.0f;
}

__global__ void count_deg(const int* __restrict__ cols, float* __restrict__ deg) {
    int t = blockIdx.x * blockDim.x + threadIdx.x;
    if (t < EE) atomicAdd(&deg[cols[t]], 1.0f);
}

__global__ void rsqrt_deg(float* __restrict__ dis) {
    int t = blockIdx.x * blockDim.x + threadIdx.x;
    if (t < NN) dis[t] = rsqrtf(dis[t]);
}

// ---------------------------------------------------------------------------
// WMMA fp32 GEMM:  Out[NN x Nreal] = A[NN x KP] * W[Kreal x Nreal]
//   - W is staged zero-padded into LDS once per block (branch-free hot loop)
//   - LDS layout is pair-major so each B fragment is one aligned ds_load_b64
//   - EMBED mode fuses the embedding gather (x == -1 -> zero row) into A
//   - Exact fp32 math via V_WMMA_F32_16X16X4_F32
// ---------------------------------------------------------------------------
template <int NT, int KP, bool EMBED>
__global__ __launch_bounds__(256) void gemm_wmma(
    const float* __restrict__ Ain, int lda,
    const int* __restrict__ xids, const float* __restrict__ emb,
    const float* __restrict__ W, int Kreal, int Nreal,
    float* __restrict__ Out, int ldc)
{
    constexpr int NP = NT * 16;
    __shared__ float Bs[KP * NP];   // element (kk,col) at ((kk>>1)*NP+col)*2 + (kk&1)

    // cooperative zero-padded stage of W into LDS
    for (int idx = threadIdx.x; idx < KP * NP; idx += 256) {
        const int kk  = idx / NP;
        const int col = idx - kk * NP;
        const float v = (kk < Kreal && col < Nreal) ? W[kk * Nreal + col] : 0.0f;
        Bs[((kk >> 1) * NP + col) * 2 + (kk & 1)] = v;
    }
    __syncthreads();

    const int lane = threadIdx.x & 31;
    const int wave = threadIdx.x >> 5;
    const int m0   = (blockIdx.x * 8 + wave) * 16;
    const int r    = lane & 15;   // row-in-tile for A, col-in-tile for B/D
    const int hi   = lane >> 4;   // K sub-group select

    float amask = 1.0f;
    long  abase;
    if (EMBED) {
        const int xi = xids[m0 + r];
        amask = (xi < 0) ? 0.0f : 1.0f;
        abase = (long)(xi < 0 ? 0 : xi) * EDIM;
    } else {
        abase = (long)(m0 + r) * lda;
    }
    const float* __restrict__ Aptr = EMBED ? emb : Ain;

    v8f acc[NT];
#pragma unroll
    for (int nt = 0; nt < NT; ++nt) acc[nt] = (v8f){};

#pragma unroll 4
    for (int k = 0; k < KP; k += 4) {
        const int kk = k + hi * 2;             // even -> 8B aligned
        v2f a = *(const v2f*)&Aptr[abase + kk];
        if (EMBED) { a.x *= amask; a.y *= amask; }
#pragma unroll
        for (int nt = 0; nt < NT; ++nt) {
            const int col = nt * 16 + r;
            const v2f b = *(const v2f*)&Bs[((kk >> 1) * NP + col) * 2];
            acc[nt] = __builtin_amdgcn_wmma_f32_16x16x4_f32(
                false, a, false, b, (short)0, acc[nt], false, false);
        }
    }

#pragma unroll
    for (int nt = 0; nt < NT; ++nt) {
        const int col = nt * 16 + r;
        if (col < Nreal) {
#pragma unroll
            for (int v = 0; v < 8; ++v) {
                const int orow = m0 + v + hi * 8;
                Out[(long)orow * ldc + col] = acc[nt][v];
            }
        }
    }
}

// ---------------------------------------------------------------------------
// edge scatter:  h[col] += dis[row]*dis[col] * xw[row]   (16 lanes per edge)
// ---------------------------------------------------------------------------
__global__ __launch_bounds__(256) void scatter_edges(
    const int* __restrict__ rows, const int* __restrict__ cols,
    const float* __restrict__ dis, const float* __restrict__ xw,
    float* __restrict__ h, int ncols, int ld)
{
    const long tid = (long)blockIdx.x * blockDim.x + threadIdx.x;
    const long e   = tid >> 4;
    const int  j0  = (int)(tid & 15);
    if (e >= (long)EE) return;
    const int rr = rows[e];
    const int cc = cols[e];
    const float nrm = dis[rr] * dis[cc];
    const float* src = xw + (long)rr * ld;
    float*       dst = h  + (long)cc * ld;
    for (int j = j0; j < ncols; j += 16)
        atomicAdd(&dst[j], nrm * src[j]);
}

// self-loop term + bias + relu (one node per block)
__global__ void finish_nodes(const float* __restrict__ dis,
                             const float* __restrict__ xw,
                             const float* __restrict__ bias,
                             float* __restrict__ h, int ncols, int ld)
{
    const int i = blockIdx.x;
    const int j = threadIdx.x;
    if (j < ncols) {
        const float d = dis[i];
        const long  o = (long)i * ld + j;
        float v = h[o] + d * d * xw[o] + bias[j];
        h[o] = v > 0.0f ? v : 0.0f;
    }
}

// gather the two global nodes of each graph -> g[512, 200]
__global__ void readout(const float* __restrict__ h, int ld, float* __restrict__ g)
{
    const int tid = blockIdx.x * blockDim.x + threadIdx.x;
    if (tid >= NUM_GRAPHS * 200) return;
    const int gi = tid / 200;
    const int t  = tid % 200;
    const int node = gi * NPG + 198 + (t >= 100 ? 1 : 0);
    const int j    = (t >= 100) ? t - 100 : t;
    g[tid] = h[(long)node * ld + j];
}

__global__ void fc1(const float* __restrict__ g, const float* __restrict__ W,
                    const float* __restrict__ b, float* __restrict__ f1)
{
    const int tid = blockIdx.x * blockDim.x + threadIdx.x;
    if (tid >= NUM_GRAPHS * 100) return;
    const int gi = tid / 100;
    const int o  = tid % 100;
    float s = b[o];
    const float* gr = g + (long)gi * 200;
    for (int k = 0; k < 200; ++k) s += gr[k] * W[k * 100 + o];
    f1[tid] = s;
}

__global__ void fc2(const float* __restrict__ f1, const float* __restrict__ W,
                    const float* __restrict__ b, float* __restrict__ out)
{
    const int tid = blockIdx.x * blockDim.x + threadIdx.x;
    if (tid >= NUM_GRAPHS * 2) return;
    const int gi = tid >> 1;
    const int o  = tid & 1;
    float s = b[o];
    const float* fr = f1 + (long)gi * 100;
    for (int k = 0; k < 100; ++k) s += fr[k] * W[k * 2 + o];
    out[tid] = s;
}

// ---------------------------------------------------------------------------
extern "C" void kernel_launch(void* const* d_in, const int* in_sizes, int n_in,
                              void* d_out, int out_size, void* d_ws, size_t ws_size,
                              hipStream_t stream)
{
    const int*   x    = (const int*)d_in[0];
    const int*   ei   = (const int*)d_in[1];   // [2, E]
    const float* emb  = (const float*)d_in[3];
    const float* W1   = (const float*)d_in[4];
    const float* b1   = (const float*)d_in[5];
    const float* W2   = (const float*)d_in[6];
    const float* b2   = (const float*)d_in[7];
    const float* W3   = (const float*)d_in[8];
    const float* b3   = (const float*)d_in[9];
    const float* fcW  = (const float*)d_in[10];
    const float* fcb  = (const float*)d_in[11];
    const float* outW = (const float*)d_in[12];
    const float* outb = (const float*)d_in[13];
    const int* rows = ei;
    const int* cols = ei + EE;

    float* ws  = (float*)d_ws;
    float* dis = ws;                          // NN
    float* XW  = dis + NN;                    // NN*112 (xw scratch, ld 64 or 112)
    float* HA  = XW + (long)NN * 112;         // NN*112 (layer1 out ld64, layer3 out ld112)
    float* HB  = HA + (long)NN * 112;         // NN*64  (layer2 out)
    float* G   = HB + (long)NN * 64;          // 512*200
    float* F1  = G + NUM_GRAPHS * 200;        // 512*100
    float* out = (float*)d_out;               // 512*2

    const int nb_n  = (NN + 255) / 256;
    const int nb_e  = (EE + 255) / 256;
    const int nb_sc = (int)(((long)EE * 16 + 255) / 256);
    const int nb_mm = NN / 128;               // 8 waves/block * 16 rows/wave

    // degrees: deg = 1 + count(col);  dis = rsqrt(deg)
    init_deg<<<nb_n, 256, 0, stream>>>(dis);
    count_deg<<<nb_e, 256, 0, stream>>>(cols, dis);
    rsqrt_deg<<<nb_n, 256, 0, stream>>>(dis);

    // ---- conv1: emb-gather @ W1 (128->50), scatter, relu -> HA (ld 64)
    gemm_wmma<4, 128, true><<<nb_mm, 256, 0, stream>>>(
        nullptr, 0, x, emb, W1, 128, 50, XW, 64);
    zero_f32<<<(int)(((long)NN * 64 + 255) / 256), 256, 0, stream>>>(HA, (long)NN * 64);
    scatter_edges<<<nb_sc, 256, 0, stream>>>(rows, cols, dis, XW, HA, 50, 64);
    finish_nodes<<<NN, 64, 0, stream>>>(dis, XW, b1, HA, 50, 64);

    // ---- conv2: HA @ W2 (50->50), scatter, relu -> HB (ld 64)
    gemm_wmma<4, 64, false><<<nb_mm, 256, 0, stream>>>(
        HA, 64, nullptr, nullptr, W2, 50, 50, XW, 64);
    zero_f32<<<(int)(((long)NN * 64 + 255) / 256), 256, 0, stream>>>(HB, (long)NN * 64);
    scatter_edges<<<nb_sc, 256, 0, stream>>>(rows, cols, dis, XW, HB, 50, 64);
    finish_nodes<<<NN, 64, 0, stream>>>(dis, XW, b2, HB, 50, 64);

    // ---- conv3: HB @ W3 (50->100), scatter, relu -> HA (ld 112)
    gemm_wmma<7, 64, false><<<nb_mm, 256, 0, stream>>>(
        HB, 64, nullptr, nullptr, W3, 50, 100, XW, 112);
    zero_f32<<<(int)(((long)NN * 112 + 255) / 256), 256, 0, stream>>>(HA, (long)NN * 112);
    scatter_edges<<<nb_sc, 256, 0, stream>>>(rows, cols, dis, XW, HA, 100, 112);
    finish_nodes<<<NN, 128, 0, stream>>>(dis, XW, b3, HA, 100, 112);

    // ---- readout + fully-connected head
    readout<<<(NUM_GRAPHS * 200 + 255) / 256, 256, 0, stream>>>(HA, 112, G);
    fc1<<<(NUM_GRAPHS * 100 + 255) / 256, 256, 0, stream>>>(G, fcW, fcb, F1);
    fc2<<<(NUM_GRAPHS * 2 + 255) / 256, 256, 0, stream>>>(F1, outW, outb, out);
}